// expressimg_21655225107033
// MI455X (gfx1250) — compile-verified
//
#include <hip/hip_runtime.h>
#include <math.h>

typedef __attribute__((ext_vector_type(2))) float v2f;
typedef __attribute__((ext_vector_type(8))) float v8f;

#define IMG 32
#define H 1024
#define WD 1024
#define WB (WD / 8)
#define HB (H / 8)
#define NBLK (WB * HB)
#define LOSS_THR 120.0f

// ---- monotone float<->int key (for integer min/max atomics) ----
__device__ __forceinline__ int fkey(float f) {
    int i = __float_as_int(f);
    return (i >= 0) ? i : (i ^ 0x7FFFFFFF);
}
__device__ __forceinline__ float fdec(int k) {
    int i = (k >= 0) ? k : (k ^ 0x7FFFFFFF);
    return __int_as_float(i);
}

// ---------------- pass 0: init workspace ----------------
__global__ void k_init(int* wsi) {
    wsi[0] = 0x7FFFFFFF;          // running min key
    wsi[1] = (int)0x80000000;     // running max key
}

// ---------------- pass 1: global min/max of delta ----------------
__global__ __launch_bounds__(256) void k_minmax(const float* __restrict__ x,
                                                int* __restrict__ wsi) {
    const long long ntot = (long long)IMG * H * WD;
    long long e = ((long long)blockIdx.x * blockDim.x + threadIdx.x) * 4;
    float mn = INFINITY, mx = -INFINITY;
    if (e < ntot) {
        const float4 v = *(const float4*)(x + e);
        long long ei = (e == 0) ? 0 : (e - 1);
        float left = x[ei];
        if ((e & (WD - 1)) == 0) left = 0.0f;   // column 0: zero-pad
        float d0 = v.x - left;
        float d1 = v.y - v.x;
        float d2 = v.z - v.y;
        float d3 = v.w - v.z;
        mn = fminf(fminf(d0, d1), fminf(d2, d3));
        mx = fmaxf(fmaxf(d0, d1), fmaxf(d2, d3));
    }
#pragma unroll
    for (int off = 16; off >= 1; off >>= 1) {
        mn = fminf(mn, __shfl_xor(mn, off, 32));
        mx = fmaxf(mx, __shfl_xor(mx, off, 32));
    }
    if ((threadIdx.x & 31) == 0) {
        atomicMin(&wsi[0], fkey(mn));
        atomicMax(&wsi[1], fkey(mx));
    }
}

// ---------------- pass 2: finalize quant params ----------------
__global__ void k_finalize(const int* __restrict__ wsi, float* __restrict__ wsf) {
    float mn = fdec(wsi[0]);
    float mx = fdec(wsi[1]);
    int neq = (mx != mn);
    float scaleq = neq ? (65535.0f / (mx - mn)) : 1.0f;
    float lsb = exp2f(rintf(log2f(mx / 32768.0f)) + 1.0f);   // part_quant1 LSB
    wsf[0] = mn;
    wsf[1] = mx;
    wsf[2] = scaleq;
    wsf[3] = 1.0f / scaleq;
    wsf[4] = lsb;
    wsf[5] = 1.0f / lsb;
    wsf[6] = neq ? 1.0f : 0.0f;
}

// ---------------- pass 3: per-8x8-block fit/quant/select ----------------
// One wave32 per block. Lane = (h = lane>>4, n = lane&15).
// Lane owns images {n, n+16}, tile rows hh in {h, h+2, h+4, h+6}
// == the V_WMMA 16x16 C/D fragment layout (p = 16*mt + 8h + v, i = 16*nt + n).
__global__ __launch_bounds__(256) void k_blocks(const float* __restrict__ x,
                                                float* __restrict__ out,
                                                const float* __restrict__ prm) {
    const int tid  = threadIdx.x;
    const int wv   = tid >> 5;
    const int lane = tid & 31;
    const int h    = lane >> 4;
    const int n    = lane & 15;
    const int blk  = blockIdx.x * 8 + wv;
    const int bw   = blk % WB;
    const int bh   = blk / WB;
    const int col0 = bw * 8;
    const int row0 = bh * 8;

    const float mn     = prm[0];
    const float scaleq = prm[2];
    const float invsq  = prm[3];
    const float lsb    = prm[4];
    const float invlsb = prm[5];
    const bool  neq    = prm[6] != 0.0f;

    // dequantized delta: part_quant round trip (identity when max==min)
    auto deq = [&](float cur, float left) -> float {
        float xd = cur - left;
        float q  = rintf((xd - mn) * scaleq) * invsq + mn;
        return neq ? q : xd;
    };

    // ---- load raw tile data: 2 images x 4 rows x 8 cols + left neighbors ----
    float raw[2][4][8];
    float lft[2][4];
#pragma unroll
    for (int nt = 0; nt < 2; ++nt) {
        const int img = n + 16 * nt;
        const float* base = x + (size_t)img * H * WD;
#pragma unroll
        for (int mt = 0; mt < 4; ++mt) {
            const int row = row0 + 2 * mt + h;
            const float* rp = base + (size_t)row * WD + col0;
            float4 lo = ((const float4*)rp)[0];
            float4 hi = ((const float4*)rp)[1];
            raw[nt][mt][0] = lo.x; raw[nt][mt][1] = lo.y;
            raw[nt][mt][2] = lo.z; raw[nt][mt][3] = lo.w;
            raw[nt][mt][4] = hi.x; raw[nt][mt][5] = hi.y;
            raw[nt][mt][6] = hi.z; raw[nt][mt][7] = hi.w;
            const float* lp = (bw == 0) ? rp : (rp - 1);   // in-bounds always
            float lval = *lp;
            lft[nt][mt] = (bw == 0) ? 0.0f : lval;
        }
    }

    // ---- publish dequantized image0/image1 columns (a1, a2) via LDS ----
    __shared__ float2 ab[8][64];   // per wave: ab[p] = (a1[p], a2[p])
    if (n < 2) {                   // lanes {0,16} hold image0, {1,17} image1
#pragma unroll
        for (int mt = 0; mt < 4; ++mt) {
            const int hh = 2 * mt + h;
            float left = lft[0][mt];
#pragma unroll
            for (int ww = 0; ww < 8; ++ww) {
                float v = deq(raw[0][mt][ww], left);
                left = raw[0][mt][ww];
                if (n == 0) ab[wv][hh * 8 + ww].x = v;
                else        ab[wv][hh * 8 + ww].y = v;
            }
        }
    }
    __syncthreads();

    // ---- single fused pass over this lane's 32 pixels:
    //      AT_A partial sums + degeneracy min/max + S = a_base @ d partials ----
    float S11 = 0.f, S12 = 0.f, S22 = 0.f, S1 = 0.f, S2 = 0.f;
    float a1mn = INFINITY, a1mx = -INFINITY, a2mn = INFINITY, a2mx = -INFINITY;
    float Sp00 = 0.f, Sp10 = 0.f, Sp20 = 0.f;
    float Sp01 = 0.f, Sp11 = 0.f, Sp21 = 0.f;
#pragma unroll
    for (int mt = 0; mt < 4; ++mt) {
        const int hh = 2 * mt + h;
        float l0 = lft[0][mt], l1 = lft[1][mt];
#pragma unroll
        for (int ww = 0; ww < 8; ++ww) {
            float2 apv = ab[wv][hh * 8 + ww];
            // AT_A partials (identical across lanes of the same half-wave)
            S11 += apv.x * apv.x; S12 += apv.x * apv.y; S22 += apv.y * apv.y;
            S1  += apv.x;         S2  += apv.y;
            a1mn = fminf(a1mn, apv.x); a1mx = fmaxf(a1mx, apv.x);
            a2mn = fminf(a2mn, apv.y); a2mx = fmaxf(a2mx, apv.y);
            // S partials for this lane's two image columns
            float d0 = deq(raw[0][mt][ww], l0); l0 = raw[0][mt][ww];
            float d1 = deq(raw[1][mt][ww], l1); l1 = raw[1][mt][ww];
            Sp00 += apv.x * d0; Sp10 += apv.y * d0; Sp20 += d0;
            Sp01 += apv.x * d1; Sp11 += apv.y * d1; Sp21 += d1;
        }
    }
    // combine even/odd-row halves (lane ^ 16 holds the other rows);
    // results are identical in both lanes of a pair -> wave-uniform sums.
    S11 += __shfl_xor(S11, 16, 32);
    S12 += __shfl_xor(S12, 16, 32);
    S22 += __shfl_xor(S22, 16, 32);
    S1  += __shfl_xor(S1,  16, 32);
    S2  += __shfl_xor(S2,  16, 32);
    a1mn = fminf(a1mn, __shfl_xor(a1mn, 16, 32));
    a1mx = fmaxf(a1mx, __shfl_xor(a1mx, 16, 32));
    a2mn = fminf(a2mn, __shfl_xor(a2mn, 16, 32));
    a2mx = fmaxf(a2mx, __shfl_xor(a2mx, 16, 32));
    float Sc00 = Sp00 + __shfl_xor(Sp00, 16, 32);
    float Sc10 = Sp10 + __shfl_xor(Sp10, 16, 32);
    float Sc20 = Sp20 + __shfl_xor(Sp20, 16, 32);
    float Sc01 = Sp01 + __shfl_xor(Sp01, 16, 32);
    float Sc11 = Sp11 + __shfl_xor(Sp11, 16, 32);
    float Sc21 = Sp21 + __shfl_xor(Sp21, 16, 32);

    const bool degen = ((a1mx - a1mn) < 1e-6f) && ((a2mx - a2mn) < 1e-6f);

    // ---- inverse of symmetric 3x3 AT_A (det==0 -> identity) ----
    const float a = S11, b = S12, c = S1, d = S22, e = S2, f = 64.0f;
    float A00 = d * f - e * e;
    float A01 = c * e - b * f;
    float A02 = b * e - d * c;
    float A11 = a * f - c * c;
    float A12 = b * c - a * e;
    float A22 = a * d - b * b;
    float det = a * A00 + b * A01 + c * A02;
    float i00, i01, i02, i11, i12, i22;
    if (det == 0.0f) {
        i00 = i11 = i22 = 1.0f; i01 = i02 = i12 = 0.0f;
    } else {
        float rd = 1.0f / det;
        i00 = A00 * rd; i01 = A01 * rd; i02 = A02 * rd;
        i11 = A11 * rd; i12 = A12 * rd; i22 = A22 * rd;
    }

    // ---- p_hat = AT_A_inv @ S (per owned image column) ----
    float P00 = i00 * Sc00 + i01 * Sc10 + i02 * Sc20;
    float P10 = i01 * Sc00 + i11 * Sc10 + i12 * Sc20;
    float P20 = i02 * Sc00 + i12 * Sc10 + i22 * Sc20;
    float P01 = i00 * Sc01 + i01 * Sc11 + i02 * Sc21;
    float P11 = i01 * Sc01 + i11 * Sc11 + i12 * Sc21;
    float P21 = i02 * Sc01 + i12 * Sc11 + i22 * Sc21;

    // ---- r = a_t(64x4) @ p_hat(4x32) via 8x V_WMMA_F32_16X16X4_F32 ----
    // A fragment (16x4): lanes 0-15 -> (K0,K1) = (a1[p], a2[p]);
    //                    lanes 16-31 -> (K2,K3) = (1, 0)
    v2f Aop[4];
#pragma unroll
    for (int mt = 0; mt < 4; ++mt) {
        float2 av = ab[wv][mt * 16 + n];
        Aop[mt].x = h ? 1.0f : av.x;
        Aop[mt].y = h ? 0.0f : av.y;
    }
    // B fragment (4x16): lanes 0-15 -> rows (K0,K1); lanes 16-31 -> (K2,K3)
    v2f Bop[2];
    Bop[0].x = h ? P20 : P00;
    Bop[0].y = h ? 0.0f : P10;
    Bop[1].x = h ? P21 : P01;
    Bop[1].y = h ? 0.0f : P11;

    v8f acc[4][2];
#pragma unroll
    for (int nt = 0; nt < 2; ++nt) {
#pragma unroll
        for (int mt = 0; mt < 4; ++mt) {
            v8f cz = {0.f, 0.f, 0.f, 0.f, 0.f, 0.f, 0.f, 0.f};
            acc[mt][nt] = __builtin_amdgcn_wmma_f32_16x16x4_f32(
                false, Aop[mt], false, Bop[nt], (short)0, cz, false, false);
        }
    }

    // ---- r1 = pow2-quantize(r); loss = sum((d1 - r1)^2) per image ----
    float lp0 = 0.f, lp1 = 0.f;
#pragma unroll
    for (int mt = 0; mt < 4; ++mt) {
        float l0 = lft[0][mt], l1 = lft[1][mt];
#pragma unroll
        for (int ww = 0; ww < 8; ++ww) {
            float d0 = deq(raw[0][mt][ww], l0); l0 = raw[0][mt][ww];
            float d1 = deq(raw[1][mt][ww], l1); l1 = raw[1][mt][ww];
            float q0 = rintf(acc[mt][0][ww] * invlsb) * lsb;
            float q1 = rintf(acc[mt][1][ww] * invlsb) * lsb;
            acc[mt][0][ww] = q0;
            acc[mt][1][ww] = q1;
            float e0 = d0 - q0; lp0 += e0 * e0;
            float e1 = d1 - q1; lp1 += e1 * e1;
        }
    }
    float loss0 = lp0 + __shfl_xor(lp0, 16, 32);
    float loss1 = lp1 + __shfl_xor(lp1, 16, 32);
    const bool sel0 = (!degen) && (loss0 <= LOSS_THR);
    const bool sel1 = (!degen) && (loss1 <= LOSS_THR);

    // ---- rr = sel ? r1 : d1 ; out = rr + x_left ----
#pragma unroll
    for (int nt = 0; nt < 2; ++nt) {
        const bool sel = nt ? sel1 : sel0;
        const int img = n + 16 * nt;
        float* ob = out + (size_t)img * H * WD;
#pragma unroll
        for (int mt = 0; mt < 4; ++mt) {
            const int row = row0 + 2 * mt + h;
            float o[8];
            float left = lft[nt][mt];
#pragma unroll
            for (int ww = 0; ww < 8; ++ww) {
                float cur = raw[nt][mt][ww];
                float d1v = deq(cur, left);
                float rr  = sel ? acc[mt][nt][ww] : d1v;
                o[ww] = rr + left;
                left = cur;
            }
            float4* op = (float4*)(ob + (size_t)row * WD + col0);
            op[0] = make_float4(o[0], o[1], o[2], o[3]);
            op[1] = make_float4(o[4], o[5], o[6], o[7]);
        }
    }
}

extern "C" void kernel_launch(void* const* d_in, const int* in_sizes, int n_in,
                              void* d_out, int out_size, void* d_ws, size_t ws_size,
                              hipStream_t stream) {
    (void)in_sizes; (void)n_in; (void)out_size; (void)ws_size;
    const float* x = (const float*)d_in[0];
    float* out = (float*)d_out;
    int*   wsi = (int*)d_ws;
    float* wsf = (float*)d_ws + 2;

    k_init<<<1, 1, 0, stream>>>(wsi);

    const long long total = (long long)IMG * H * WD;      // 33,554,432
    const int mm_threads = 256;
    const int mm_blocks  = (int)((total / 4 + mm_threads - 1) / mm_threads); // 32768
    k_minmax<<<mm_blocks, mm_threads, 0, stream>>>(x, wsi);

    k_finalize<<<1, 1, 0, stream>>>(wsi, wsf);

    k_blocks<<<NBLK / 8, 256, 0, stream>>>(x, out, wsf);  // 2048 WGs x 8 waves
}